// ForwardKinematicsLayer_9225589752520
// MI455X (gfx1250) — compile-verified
//
#include <hip/hip_runtime.h>

// CDNA5 / gfx1250. Wave32. fp32 WMMA 16x16x4 drives the 4x4 transform chain
// (4 independent batch chains block-diagonally packed per wave).
// Full 4x4 matrices kept in LDS (row3 = (0,0,0,1) physical) so WMMA operand
// builds are pure loads and the transforms output is a linear B128 copy.

typedef float v2f __attribute__((ext_vector_type(2)));
typedef float v8f __attribute__((ext_vector_type(8)));

#define B_TOTAL 262144
#define NJ 24
#define WAVES_PER_BLOCK 4

// per-wave LDS layout (in floats)
#define ROT_F   (4 * NJ * 6)     // 576  staged rotations; later reused as dummy sink
#define POS_F   (4 * NJ * 3)     // 288  staged positions
#define LMAT_F  (4 * NJ * 16)    // 1536 local 4x4 transforms (row3 explicit)
#define G_F     (4 * NJ * 16)    // 1536 global 4x4 transforms (== output layout)
#define WAVE_F  (ROT_F + POS_F + LMAT_F + G_F)   // 3936 floats = 15744 B / wave

__global__ __launch_bounds__(32 * WAVES_PER_BLOCK)
void fk_wmma_kernel(const float* __restrict__ rot6d,
                    const float* __restrict__ positions,
                    float* __restrict__ out) {
  __shared__ float smem[WAVE_F * WAVES_PER_BLOCK];   // 62976 B for 4 waves

  const int lane = threadIdx.x & 31;
  const int wid  = threadIdx.x >> 5;
  const int wave = blockIdx.x * WAVES_PER_BLOCK + wid;
  const int b0   = wave * 4;                     // 4 batch elements per wave

  float* sRot = smem + wid * WAVE_F;             // dead after conversions -> dummy sink
  float* sPos = sRot + ROT_F;
  float* sL   = sPos + POS_F;                    // sL[(b*NJ+j)*16 + r*4 + c]
  float* sG   = sL + LMAT_F;                     // sG[(b*NJ+j)*16 + r*4 + c]

  // ---- stage inputs coalesced into LDS via B128 ----
  {
    const float4* r4 = (const float4*)(rot6d     + (size_t)b0 * (NJ * 6)); // 144 q
    const float4* p4 = (const float4*)(positions + (size_t)b0 * (NJ * 3)); //  72 q
    for (int q = lane; q < ROT_F / 4; q += 32) *(float4*)(sRot + 4 * q) = r4[q];
    for (int q = lane; q < POS_F / 4; q += 32) *(float4*)(sPos + 4 * q) = p4[q];
  }
  asm volatile("s_wait_dscnt 0" ::: "memory");   // intra-wave LDS visibility

  // ---- rot6d -> full 4x4 local transforms: 96 jobs per wave, 3 per lane ----
  #pragma unroll
  for (int t = 0; t < 3; ++t) {
    const int job = lane + t * 32;
    const int b = job / NJ;
    const int j = job - b * NJ;
    const float* r6 = sRot + (b * NJ + j) * 6;
    float a1x = r6[0], a1y = r6[1], a1z = r6[2];
    float a2x = r6[3], a2y = r6[4], a2z = r6[5];
    float n1  = sqrtf(a1x * a1x + a1y * a1y + a1z * a1z);
    float b1x = a1x / n1, b1y = a1y / n1, b1z = a1z / n1;
    float d   = b1x * a2x + b1y * a2y + b1z * a2z;
    float u2x = a2x - d * b1x, u2y = a2y - d * b1y, u2z = a2z - d * b1z;
    float n2  = sqrtf(u2x * u2x + u2y * u2y + u2z * u2z);
    float b2x = u2x / n2, b2y = u2y / n2, b2z = u2z / n2;
    float b3x = b1y * b2z - b1z * b2y;
    float b3y = b1z * b2x - b1x * b2z;
    float b3z = b1x * b2y - b1y * b2x;
    const float* p3 = sPos + (b * NJ + j) * 3;
    float* L = sL + (b * NJ + j) * 16;           // 64B aligned
    *(float4*)(L +  0) = make_float4(b1x, b1y, b1z, p3[0]);
    *(float4*)(L +  4) = make_float4(b2x, b2y, b2z, p3[1]);
    *(float4*)(L +  8) = make_float4(b3x, b3y, b3z, p3[2]);
    *(float4*)(L + 12) = make_float4(0.f, 0.f, 0.f, 1.f);
  }

  // ---- g[b][0] = L[b][0] : 16 float4 copies ----
  if (lane < 16) {
    int b = lane >> 2, r = lane & 3;
    int off = b * NJ * 16 + r * 4;
    *(float4*)(sG + off) = *(const float4*)(sL + off);
  }
  asm volatile("s_wait_dscnt 0" ::: "memory");

  // ---- WMMA chain: D(16x16) = A(16x4:stacked L_i^T) x B(4x16:[g_p^T x4]) ----
  // A vgpr j : lane%16 = M (M=4b+r'), K = j + 2*(lane>=16) ; A[M,K] = L_b[K, r']
  // B vgpr j : lane%16 = N (N=4b+c'), K = j + 2*(lane>=16) ; B[K,N] = g_b[c', K]
  // D vgpr v : M = v + 8*(lane>=16), N = lane%16 ; diag block = newg_b^T
  const int idx  = lane & 15;
  const int half = lane >> 4;
  const int bb   = idx >> 2;    // which of the 4 batches this lane serves
  const int sub  = idx & 3;     // r' for A loads, c'(= output row) for B/scatter

  // scatter routing: lane's diagonal data sits in D quad (v=0..3) iff bb even,
  // quad (v=4..7) iff bb odd; only lanes with bb>>1 == half carry batch bb.
  const bool active = ((bb >> 1) == half);
  const bool sel_lo = active && ((bb & 1) == 0);
  const bool sel_hi = active && ((bb & 1) == 1);
  float* dummyA = sRot + 8 * lane;           // dead staging area as sink
  float* dummyB = dummyA + 4;

  constexpr int PAR[NJ] = {0,0,0,0,1,2,3,4,5,6,7,8,9,9,9,12,13,14,16,17,18,19,20,21};

  #pragma unroll
  for (int i = 1; i < NJ; ++i) {
    const int p = PAR[i];
    v2f A, Bm;
    #pragma unroll
    for (int jv = 0; jv < 2; ++jv) {
      const int K = jv + 2 * half;                       // 0..3, all physical now
      A[jv]  = sL[(bb * NJ + i) * 16 + K * 4 + sub];     // L_b[K, r']
      Bm[jv] = sG[(bb * NJ + p) * 16 + sub * 4 + K];     // g_b[c', K] (b64 pair)
    }
    v8f C = {0.f,0.f,0.f,0.f,0.f,0.f,0.f,0.f};
    v8f D = __builtin_amdgcn_wmma_f32_16x16x4_f32(
        /*neg_a=*/false, A, /*neg_b=*/false, Bm,
        /*c_mod=*/(short)0, C, /*reuse_a=*/false, /*reuse_b=*/false);

    // branch-free scatter: write BOTH quads as float4; at most one pointer is
    // real per lane (row 'sub' of newg_bb), any other goes to the dummy sink.
    float* realp = sG + (bb * NJ + i) * 16 + sub * 4;    // 16B aligned
    float* lo = sel_lo ? realp : dummyA;
    float* hi = sel_hi ? realp : dummyB;
    *(float4*)lo = make_float4(D[0], D[1], D[2], D[3]);
    *(float4*)hi = make_float4(D[4], D[5], D[6], D[7]);
    asm volatile("s_wait_dscnt 0" ::: "memory");
  }

  // ---- outputs: coordinates (B,24,3) then transforms (B,24,4,4) ----
  // coordinates: 288 scalars per wave, coalesced b32 (w == 1 exactly)
  float* cbase = out + (size_t)b0 * (NJ * 3);
  for (int f = lane; f < 4 * NJ * 3; f += 32) {
    int b = f / (NJ * 3); int rem = f - b * (NJ * 3);
    int j = rem / 3;      int r   = rem - j * 3;
    cbase[f] = sG[(b * NJ + j) * 16 + r * 4 + 3];
  }
  // transforms: sG layout == output layout -> straight linear B128 copy
  float4* tbase = (float4*)(out + (size_t)B_TOTAL * (NJ * 3) + (size_t)b0 * (NJ * 16));
  #pragma unroll
  for (int it = 0; it < 12; ++it) {
    int q = lane + it * 32;                  // 384 float4 per wave, exact
    tbase[q] = *(const float4*)(sG + 4 * q);
  }
}

extern "C" void kernel_launch(void* const* d_in, const int* in_sizes, int n_in,
                              void* d_out, int out_size, void* d_ws, size_t ws_size,
                              hipStream_t stream) {
  const float* rot = (const float*)d_in[0];   // (B, 24, 6) fp32
  const float* pos = (const float*)d_in[1];   // (B, 24, 3) fp32
  float* out = (float*)d_out;                 // coords (B,24,3) ++ transforms (B,24,4,4)
  (void)in_sizes; (void)n_in; (void)out_size; (void)d_ws; (void)ws_size;

  dim3 block(32 * WAVES_PER_BLOCK);                       // 128 threads = 4 waves
  dim3 grid(B_TOTAL / (4 * WAVES_PER_BLOCK));             // 4 batches per wave
  hipLaunchKernelGGL(fk_wmma_kernel, grid, block, 0, stream, rot, pos, out);
}